// SSD_15126874817008
// MI455X (gfx1250) — compile-verified
//
#include <hip/hip_runtime.h>
#include <math.h>

// ---------------- problem constants (from reference) ----------------
#define BATCH   128
#define NPRIOR  8732
#define NCLASS  21
#define NCLS    20          // foreground classes
#define TOPK    200
#define KEEP    200
#define CONF_T  0.01f
#define NMS_T   0.45f
#define VAR0    0.1f
#define VAR1    0.2f

typedef __attribute__((ext_vector_type(2))) float v2f;
typedef __attribute__((ext_vector_type(8))) float v8f;

__device__ __forceinline__ float clamp01(float x) {
    return fminf(fmaxf(x, 0.0f), 1.0f);
}

// Async global->LDS copy of one b32 per lane (CDNA5, ASYNCcnt-tracked).
// dst_lds_byte_off = per-lane byte offset into workgroup LDS,
// gaddr            = per-lane 64-bit global address.
__device__ __forceinline__ void async_copy_b32(unsigned dst_lds_byte_off,
                                               unsigned long long gaddr) {
    asm volatile("global_load_async_to_lds_b32 %0, %1, off"
                 :: "v"(dst_lds_byte_off), "v"(gaddr)
                 : "memory");
}
__device__ __forceinline__ void wait_asynccnt0() {
    asm volatile("s_wait_asynccnt 0x0" ::: "memory");
}

// ---------------------------------------------------------------
// K0: decode boxes.  one thread per (b, prior). float4 in/out.
// ---------------------------------------------------------------
__global__ __launch_bounds__(256)
void ssd_decode_kernel(const float* __restrict__ loc,
                       const float* __restrict__ dbox,
                       float* __restrict__ boxes, int total) {
    int i = blockIdx.x * blockDim.x + threadIdx.x;
    if (i >= total) return;
    int n = i % NPRIOR;
    float4 l = ((const float4*)loc)[i];
    float4 d = ((const float4*)dbox)[n];
    float cx = d.x + l.x * VAR0 * d.z;
    float cy = d.y + l.y * VAR0 * d.w;
    float w  = d.z * __expf(l.z * VAR1);
    float h  = d.w * __expf(l.w * VAR1);
    float x1 = cx - 0.5f * w;
    float y1 = cy - 0.5f * h;
    float4 o;
    o.x = clamp01(x1);
    o.y = clamp01(y1);
    o.z = clamp01(x1 + w);
    o.w = clamp01(y1 + h);
    ((float4*)boxes)[i] = o;
}

// ---------------------------------------------------------------
// K1: softmax over 21 classes; WMMA (f32 16x16x4) computes the
// per-prior exp row-sums for 16 priors per wave.  Writes class-major
// score planes scoresT[b][c-1][n] for c = 1..20.
// ---------------------------------------------------------------
#define WAVES 8
#define PPW   16            // priors per wave
#define PPB   (WAVES * PPW) // 128 priors per block

__global__ __launch_bounds__(256)
void ssd_softmax_kernel(const float* __restrict__ conf,
                        float* __restrict__ scoresT) {
    __shared__ float e[WAVES][PPW][24];     // exp tile, classes padded to 24
    __shared__ float pmax[WAVES][PPW][2];
    __shared__ float psum[WAVES][PPW];

    const int b    = blockIdx.y;
    const int wave = threadIdx.x >> 5;
    const int lane = threadIdx.x & 31;
    const int m    = lane & 15;             // prior row within wave tile
    const int half = lane >> 4;             // lane half (K split per ISA A-layout)
    const int prior = blockIdx.x * PPB + wave * PPW + m;
    const bool valid = prior < NPRIOR;

    const float* cp = conf + ((size_t)b * NPRIOR + (valid ? prior : 0)) * NCLASS;
    __builtin_prefetch(cp + (size_t)PPB * NCLASS, 0, 0);  // next tile of the 94MB stream
    const int c0 = half ? 11 : 0;           // half 0: classes 0..10, half 1: 11..20

    float v[11];
    float mx = -3.0e38f;
#pragma unroll
    for (int j = 0; j < 11; ++j) {
        int c = c0 + j;
        float x = (valid && c < NCLASS) ? cp[c] : -3.0e38f;
        v[j] = x;
        mx = fmaxf(mx, x);
    }
    pmax[wave][m][half] = mx;
    __syncthreads();
    float rmx = fmaxf(pmax[wave][m][0], pmax[wave][m][1]);

#pragma unroll
    for (int j = 0; j < 11; ++j) {
        int c = c0 + j;
        if (c < NCLASS) e[wave][m][c] = __expf(v[j] - rmx);
    }
    if (half) { e[wave][m][21] = 0.0f; e[wave][m][22] = 0.0f; e[wave][m][23] = 0.0f; }
    __syncthreads();

    // Row sums via D = E(16x24) * Ones : six accumulating f32 WMMAs.
    // A layout (ISA 16x4 f32): lanes 0-15 rows, VGPR0/1 = K+0/K+1; lanes 16-31 = K+2/K+3.
    v8f acc = {0.f, 0.f, 0.f, 0.f, 0.f, 0.f, 0.f, 0.f};
    v2f bones = {1.0f, 1.0f};               // all-ones B is layout-invariant
#pragma unroll
    for (int k = 0; k < 6; ++k) {
        int kb = 4 * k + 2 * half;
        v2f a = {e[wave][m][kb + 0], e[wave][m][kb + 1]};
        acc = __builtin_amdgcn_wmma_f32_16x16x4_f32(
            /*neg_a=*/false, a, /*neg_b=*/false, bones,
            /*c_mod=*/(short)0, acc, /*reuse_a=*/false, /*reuse_b=*/false);
    }
    // C/D layout: VGPR r = row r (lanes 0-15) / row r+8 (lanes 16-31); all cols equal.
    if (lane == 0) {
#pragma unroll
        for (int r = 0; r < 8; ++r) psum[wave][r] = acc[r];
    }
    if (lane == 16) {
#pragma unroll
        for (int r = 0; r < 8; ++r) psum[wave][r + 8] = acc[r];
    }
    __syncthreads();

    float inv = 1.0f / psum[wave][m];
#pragma unroll
    for (int j = 0; j < 11; ++j) {
        int c = c0 + j;
        if (c >= 1 && c < NCLASS && valid) {
            scoresT[((size_t)b * NCLS + (c - 1)) * NPRIOR + prior] = e[wave][m][c] * inv;
        }
    }
}

// ---------------------------------------------------------------
// K2: per-(b,class) top-200 selection.  Score plane async-staged to
// LDS (35 KB of the 320 KB WGP LDS) via GLOBAL_LOAD_ASYNC_TO_LDS_B32,
// then 200 stable block-argmax rounds entirely in LDS.
// ---------------------------------------------------------------
__global__ __launch_bounds__(256)
void ssd_topk_kernel(const float* __restrict__ scoresT,
                     const float* __restrict__ boxes,
                     float* __restrict__ topS,
                     float* __restrict__ topB) {
    __shared__ float ls[NPRIOR];
    __shared__ float rmax[256];
    __shared__ int   ridx[256];

    const int bc = blockIdx.x;
    const int b  = bc / NCLS;
    const int t  = threadIdx.x;
    const float* s  = scoresT + (size_t)bc * NPRIOR;
    const float* bx = boxes + (size_t)b * NPRIOR * 4;

    // async stage: memory -> LDS, no VGPR data path
    for (int i = t; i < NPRIOR; i += 256) {
        async_copy_b32((unsigned)(unsigned long long)&ls[i],
                       (unsigned long long)(s + i));
    }
    wait_asynccnt0();
    __syncthreads();
    // in-place threshold pass (claimed entries later become -3, below the -1 fill)
    for (int i = t; i < NPRIOR; i += 256) {
        float v = ls[i];
        ls[i] = (v > CONF_T) ? v : -1.0f;
    }
    __syncthreads();

    float* oS = topS + (size_t)bc * TOPK;
    float* oB = topB + (size_t)bc * TOPK * 4;

    for (int k = 0; k < TOPK; ++k) {
        float m = -3.0f; int mi = 0;
        for (int i = t; i < NPRIOR; i += 256) {
            float v = ls[i];
            if (v > m) { m = v; mi = i; }
        }
        rmax[t] = m; ridx[t] = mi;
        __syncthreads();
        for (int off = 128; off > 0; off >>= 1) {
            if (t < off) {
                float o = rmax[t + off]; int oi = ridx[t + off];
                if (o > rmax[t] || (o == rmax[t] && oi < ridx[t])) { rmax[t] = o; ridx[t] = oi; }
            }
            __syncthreads();
        }
        int   wi = ridx[0];
        float wv = rmax[0];
        if (t == 0) { oS[k] = wv; ls[wi] = -3.0f; }   // claim; below the -1 fill
        if (t < 4)  oB[k * 4 + t] = bx[(size_t)wi * 4 + t];
        __syncthreads();
    }
}

// ---------------------------------------------------------------
// K3: greedy NMS per (b,class); 200 candidates in LDS, 200-wide
// parallel suppression per step.  NaN IoU suppresses (matches torch).
// ---------------------------------------------------------------
__global__ __launch_bounds__(256)
void ssd_nms_kernel(float* __restrict__ topS, float* __restrict__ topB) {
    __shared__ float s[TOPK], x1[TOPK], y1[TOPK], x2[TOPK], y2[TOPK], ar[TOPK];
    __shared__ int alive[TOPK];

    const int bc = blockIdx.x;
    const int t  = threadIdx.x;
    float* S  = topS + (size_t)bc * TOPK;
    float* Bx = topB + (size_t)bc * TOPK * 4;

    if (t < TOPK) {
        float sv = S[t];
        float a = Bx[t * 4 + 0], b = Bx[t * 4 + 1];
        float c = Bx[t * 4 + 2], d = Bx[t * 4 + 3];
        s[t] = sv; x1[t] = a; y1[t] = b; x2[t] = c; y2[t] = d;
        ar[t] = (c - a) * (d - b);
        alive[t] = (sv > CONF_T) ? 1 : 0;
    }
    __syncthreads();

    for (int i = 0; i < TOPK; ++i) {
        bool take = alive[i] != 0;
        if (take && t < TOPK && t > i && alive[t]) {
            float xx1 = fmaxf(x1[t], x1[i]);
            float yy1 = fmaxf(y1[t], y1[i]);
            float xx2 = fminf(x2[t], x2[i]);
            float yy2 = fminf(y2[t], y2[i]);
            float inter = fmaxf(xx2 - xx1, 0.0f) * fmaxf(yy2 - yy1, 0.0f);
            float iou = inter / (ar[t] + ar[i] - inter);
            if (!(iou <= NMS_T)) alive[t] = 0;    // NaN -> suppressed
        }
        __syncthreads();
    }

    if (t < TOPK) {
        bool keep = (alive[t] != 0) && (s[t] > CONF_T);
        S[t] = keep ? s[t] : 0.0f;
        if (!keep) {
            Bx[t * 4 + 0] = 0.0f; Bx[t * 4 + 1] = 0.0f;
            Bx[t * 4 + 2] = 0.0f; Bx[t * 4 + 3] = 0.0f;
        }
    }
}

// ---------------------------------------------------------------
// K4: per-batch global top-200 over 4000 candidates + per-class
// stable compaction into out[b][21][200][5].  Candidate scores are
// async-staged to LDS.
// ---------------------------------------------------------------
#define NCAND (NCLS * TOPK)   // 4000

__global__ __launch_bounds__(256)
void ssd_emit_kernel(const float* __restrict__ topS,
                     const float* __restrict__ topB,
                     float* __restrict__ out) {
    __shared__ float fs[NCAND];
    __shared__ float rmax[256];
    __shared__ int   ridx[256];
    __shared__ int   gI[KEEP];
    __shared__ float gV[KEEP];

    const int b = blockIdx.x;
    const int t = threadIdx.x;
    const float* S  = topS + (size_t)b * NCAND;
    const float* Bx = topB + (size_t)b * NCAND * 4;
    float* o = out + (size_t)b * NCLASS * KEEP * 5;

    // async stage candidate scores (raw copy; claimed entries get -2, values >= 0)
    for (int i = t; i < NCAND; i += 256) {
        async_copy_b32((unsigned)(unsigned long long)&fs[i],
                       (unsigned long long)(S + i));
    }
    for (int i = t; i < NCLASS * KEEP * 5; i += 256) o[i] = 0.0f;  // zero slab
    wait_asynccnt0();
    __syncthreads();

    for (int k = 0; k < KEEP; ++k) {
        float m = -2.0f; int mi = 0;
        for (int i = t; i < NCAND; i += 256) {
            float v = fs[i];
            if (v > m) { m = v; mi = i; }
        }
        rmax[t] = m; ridx[t] = mi;
        __syncthreads();
        for (int off = 128; off > 0; off >>= 1) {
            if (t < off) {
                float ov = rmax[t + off]; int oi = ridx[t + off];
                if (ov > rmax[t] || (ov == rmax[t] && oi < ridx[t])) { rmax[t] = ov; ridx[t] = oi; }
            }
            __syncthreads();
        }
        if (t == 0) { gI[k] = ridx[0]; gV[k] = rmax[0]; fs[ridx[0]] = -2.0f; }
        __syncthreads();
    }

    // stable per-class compaction (one thread per foreground class)
    if (t < NCLS) {
        int cl = t;          // label = cl + 1
        int slot = 0;
        for (int k = 0; k < KEEP; ++k) {
            int wi = gI[k]; float wv = gV[k];
            if (wv > 0.0f && (wi / TOPK) == cl) {
                float* row = o + ((size_t)(cl + 1) * KEEP + slot) * 5;
                row[0] = wv;
                row[1] = Bx[(size_t)wi * 4 + 0];
                row[2] = Bx[(size_t)wi * 4 + 1];
                row[3] = Bx[(size_t)wi * 4 + 2];
                row[4] = Bx[(size_t)wi * 4 + 3];
                ++slot;
            }
        }
    }
}

// ---------------------------------------------------------------
extern "C" void kernel_launch(void* const* d_in, const int* in_sizes, int n_in,
                              void* d_out, int out_size, void* d_ws, size_t ws_size,
                              hipStream_t stream) {
    (void)in_sizes; (void)n_in; (void)out_size; (void)ws_size;
    const float* loc  = (const float*)d_in[0];   // [B,N,4]
    const float* conf = (const float*)d_in[1];   // [B,N,21]
    const float* dbox = (const float*)d_in[2];   // [N,4]
    float* out = (float*)d_out;                  // [B,21,200,5]

    // workspace layout (f32): boxes | scoresT | topS | topB  (~118 MB)
    float* boxes   = (float*)d_ws;
    float* scoresT = boxes   + (size_t)BATCH * NPRIOR * 4;
    float* topS    = scoresT + (size_t)BATCH * NCLS * NPRIOR;
    float* topB    = topS    + (size_t)BATCH * NCLS * TOPK;

    const int total = BATCH * NPRIOR;
    ssd_decode_kernel<<<(total + 255) / 256, 256, 0, stream>>>(loc, dbox, boxes, total);

    dim3 g1((NPRIOR + PPB - 1) / PPB, BATCH);
    ssd_softmax_kernel<<<g1, 256, 0, stream>>>(conf, scoresT);

    ssd_topk_kernel<<<BATCH * NCLS, 256, 0, stream>>>(scoresT, boxes, topS, topB);
    ssd_nms_kernel <<<BATCH * NCLS, 256, 0, stream>>>(topS, topB);
    ssd_emit_kernel<<<BATCH, 256, 0, stream>>>(topS, topB, out);
}